// CubeForwardProjection_5325759447782
// MI455X (gfx1250) — compile-verified
//
#include <hip/hip_runtime.h>
#include <hip/hip_bf16.h>
#include <stdint.h>

typedef __attribute__((ext_vector_type(2))) float v2f;
typedef __attribute__((ext_vector_type(8))) float v8f;

static constexpr int NB = 2, SSEG = 8, CCH = 128;
static constexpr int HQ = 120, WQ = 160, HF = 480, WF = 640;
static constexpr int TS = 32;           // output tile size T
static constexpr int SMAX = 160;        // max crop side rounded up to 16

// workspace layout (bytes)
static constexpr size_t WS_META_OFF  = 0;                       // 16 x 4 ints
static constexpr size_t WS_MASKQ_OFF = 1024;                    // 16 x 19200 u8
static constexpr size_t WS_CNT_OFF   = 1024 + 16 * (size_t)HQ * WQ + 768; // 16 x 32768 f32 (16B aligned: 308992)

// ---------------------------------------------------------------- zero fill
__global__ void k_zero_f4(float4* __restrict__ p, long n4) {
  long i = (long)blockIdx.x * blockDim.x + threadIdx.x;
  long stride = (long)gridDim.x * blockDim.x;
  float4 z = make_float4(0.f, 0.f, 0.f, 0.f);
  for (; i < n4; i += stride) p[i] = z;
}

// ------------------------------------------------- mask meta + quarter mask
__global__ void __launch_bounds__(256) k_meta(const unsigned char* __restrict__ masks,
                                              const float* __restrict__ depths,
                                              int* __restrict__ meta,
                                              unsigned char* __restrict__ maskq) {
  int bs = blockIdx.x;                 // 0..15
  int b  = bs / SSEG;
  int tid = threadIdx.x;
  const unsigned char* mk = masks + (size_t)bs * HF * WF;
  const float*         dp = depths + (size_t)b * HF * WF;
  unsigned char*       mq = maskq + (size_t)bs * HQ * WQ;

  int cnt = 0, ymin = 0x7fffffff, ymax = -1, xmin = 0x7fffffff, xmax = -1;
  for (int i = tid; i < HQ * WQ; i += 256) {
    int sy = i / WQ, sx = i - sy * WQ;
    int fi = (sy * 4) * WF + sx * 4;
    int m = (mk[fi] != 0) && (dp[fi] > 0.0f);
    mq[i] = (unsigned char)m;
    if (m) {
      cnt++;
      ymin = min(ymin, sy); ymax = max(ymax, sy);
      xmin = min(xmin, sx); xmax = max(xmax, sx);
    }
  }
  __shared__ int sc[256], sy0[256], sy1[256], sx0[256], sx1[256];
  sc[tid] = cnt; sy0[tid] = ymin; sy1[tid] = ymax; sx0[tid] = xmin; sx1[tid] = xmax;
  __syncthreads();
  for (int off = 128; off > 0; off >>= 1) {
    if (tid < off) {
      sc[tid] += sc[tid + off];
      sy0[tid] = min(sy0[tid], sy0[tid + off]);
      sy1[tid] = max(sy1[tid], sy1[tid + off]);
      sx0[tid] = min(sx0[tid], sx0[tid + off]);
      sx1[tid] = max(sx1[tid], sx1[tid + off]);
    }
    __syncthreads();
  }
  if (tid == 0) {
    int c = sc[0];
    int side = 0, ny0 = 0, nx0 = 0;
    if (c >= 10) {
      int y0 = sy0[0], y1 = sy1[0], x0 = sx0[0], x1 = sx1[0];
      side = max(y1 - y0, x1 - x0);
      int cy = (y0 + y1) >> 1, cx = (x0 + x1) >> 1;
      ny0 = max(cy - (side >> 1), 0);
      nx0 = max(cx - (side >> 1), 0);
    }
    meta[bs * 4 + 0] = c;  meta[bs * 4 + 1] = ny0;
    meta[bs * 4 + 2] = nx0; meta[bs * 4 + 3] = side;
  }
}

// ---------------------------------- bilinear resize: f2 = Wy @ (img @ Wx^T)
// one block per (b,s,channel); x-direction = exact 2-tap gather into LDS,
// y-direction = dense f32 GEMM on V_WMMA_F32_16X16X4_F32 (exact in f32).
__global__ void __launch_bounds__(128) k_resize(const float* __restrict__ feat,
                                                const int* __restrict__ meta,
                                                const unsigned char* __restrict__ maskq,
                                                float* __restrict__ f2) {
  int blk = blockIdx.x;
  int c  = blk % CCH;
  int bs = blk / CCH;
  int b  = bs / SSEG;

  if (meta[bs * 4 + 0] < 10) return;   // stays zero (uniform per block)
  int ny0 = meta[bs * 4 + 1], nx0 = meta[bs * 4 + 2], side = meta[bs * 4 + 3];
  int S16 = (side + 15) & ~15;         // K dimension, padded; <= SMAX

  __shared__ float Wy[TS][SMAX];       // 20 KB
  __shared__ float tmp1[SMAX][TS + 1]; // 21 KB (pad avoids bank conflicts)

  int tid = threadIdx.x;
  // zero the used part of Wy
  for (int i = tid; i < TS * S16; i += 128) Wy[i / S16][i % S16] = 0.0f;
  __syncthreads();
  // build sparse y-interp matrix (2 nonzeros per row; += handles lo==hi clip)
  if (tid < TS) {
    float y = (tid + 0.5f) * (side * 0.03125f) - 0.5f;   // exact in f32
    y = fminf(fmaxf(y, 0.0f), (float)(side - 1));
    int lo = (int)floorf(y);
    int hi = min(lo + 1, side - 1);
    float f = y - (float)lo;
    Wy[tid][lo] += 1.0f - f;
    Wy[tid][hi] += f;
  }
  // stage 1: x-direction bilinear of masked crop -> tmp1[S16][32]
  const float* fc = feat + ((size_t)b * CCH + c) * (HQ * WQ);
  const unsigned char* mq = maskq + (size_t)bs * (HQ * WQ);
  int tx = tid & 31;
  int r0 = tid >> 5;                   // 4 rows per pass
  float x = (tx + 0.5f) * (side * 0.03125f) - 0.5f;
  x = fminf(fmaxf(x, 0.0f), (float)(side - 1));
  int xlo = (int)floorf(x);
  int xhi = min(xlo + 1, side - 1);
  float fx = x - (float)xlo;
  int sxa = nx0 + xlo, sxb = nx0 + xhi;
  for (int r = r0; r < S16; r += 4) {
    float v = 0.0f;
    int sy = ny0 + r;
    if (r < side && sy < HQ) {         // crop pads with zeros beyond image
      float va = 0.0f, vb = 0.0f;
      if (sxa < WQ && mq[sy * WQ + sxa]) va = fc[sy * WQ + sxa];
      if (sxb < WQ && mq[sy * WQ + sxb]) vb = fc[sy * WQ + sxb];
      v = va * (1.0f - fx) + vb * fx;
    }
    tmp1[r][tx] = v;
  }
  __syncthreads();

  // stage 2: out(32x32) = Wy(32xS16) @ tmp1(S16x32); 4 waves = 4 16x16 tiles
  int wave = tid >> 5;
  int lane = tid & 31;
  int mi = (wave >> 1) * 16;           // output row tile base
  int ni = (wave & 1) * 16;            // output col tile base
  int half = lane >> 4;                // ISA layout: lanes 16-31 hold K+2,K+3
  int l = lane & 15;
  v8f acc = {};
  for (int k = 0; k < S16; k += 4) {
    int ka = k + half * 2;
    v2f a, bf;
    a.x  = Wy[mi + l][ka];
    a.y  = Wy[mi + l][ka + 1];
    bf.x = tmp1[ka][ni + l];
    bf.y = tmp1[ka + 1][ni + l];
    acc = __builtin_amdgcn_wmma_f32_16x16x4_f32(false, a, false, bf,
                                                (short)0, acc, false, false);
  }
  // D layout: VGPR j -> (M=j, N=lane) lanes 0-15; (M=j+8, N=lane-16) lanes 16-31
  float* outp = f2 + ((size_t)bs * CCH + c) * (TS * TS);
  int col = ni + l;
  int rbase = mi + half * 8;
#pragma unroll
  for (int j = 0; j < 8; ++j) outp[(rbase + j) * TS + col] = acc[j];
}

// ------------------------------------------------------ pixelate (scatter)
__global__ void __launch_bounds__(256) k_pixelate(const float* __restrict__ depths,
                                                  const float* __restrict__ intr,
                                                  const int* __restrict__ meta,
                                                  const unsigned char* __restrict__ maskq,
                                                  const float* __restrict__ f2,
                                                  float* __restrict__ g,
                                                  float* __restrict__ cntbuf) {
  int bs = blockIdx.x;
  int b  = bs / SSEG;
  if (meta[bs * 4 + 0] < 10) return;
  int ny0 = meta[bs * 4 + 1], nx0 = meta[bs * 4 + 2], side = meta[bs * 4 + 3];
  int tid = threadIdx.x;

  __shared__ float px[1024], py[1024], pz[1024];
  __shared__ int   sflat[1024];
  __shared__ unsigned char sw[1024];

  const float* dp = depths + (size_t)b * HF * WF;
  const unsigned char* mq = maskq + (size_t)bs * HQ * WQ;
  float fxk = intr[b * 9 + 0], cxk = intr[b * 9 + 2];
  float fyk = intr[b * 9 + 4], cyk = intr[b * 9 + 5];
  float kz  = intr[b * 9 + 8] * 4.0f;  // K[2,2] *= 4 in reference

  // nearest-resized, masked, cropped back-projected points
  for (int i = tid; i < 1024; i += 256) {
    int ty = i >> 5, txx = i & 31;
    int yi = (ty * side) >> 5;         // == floor(ty*side/32), exact
    int xi = (txx * side) >> 5;
    int sy = ny0 + yi, sx = nx0 + xi;
    float X = 0.f, Y = 0.f, Z = 0.f;
    if (sy < HQ && sx < WQ && mq[sy * WQ + sx]) {
      float d = dp[(sy * 4) * WF + sx * 4];
      X = (kz * (float)sx - cxk) / fxk * d;   // cam/cam_z * depth
      Y = (kz * (float)sy - cyk) / fyk * d;
      Z = d;
    }
    px[i] = X; py[i] = Y; pz[i] = Z;
  }
  __syncthreads();

  // min/max over valid points (all three coords nonzero)
  float mnx = 1e30f, mny = 1e30f, mnz = 1e30f;
  float mxx = -1e30f, mxy = -1e30f, mxz = -1e30f;
  for (int i = tid; i < 1024; i += 256) {
    if (px[i] != 0.f && py[i] != 0.f && pz[i] != 0.f) {
      mnx = fminf(mnx, px[i]); mny = fminf(mny, py[i]); mnz = fminf(mnz, pz[i]);
      mxx = fmaxf(mxx, px[i]); mxy = fmaxf(mxy, py[i]); mxz = fmaxf(mxz, pz[i]);
    }
  }
  __shared__ float r0[256], r1[256], r2[256], r3[256], r4[256], r5[256];
  r0[tid] = mnx; r1[tid] = mny; r2[tid] = mnz;
  r3[tid] = mxx; r4[tid] = mxy; r5[tid] = mxz;
  __syncthreads();
  for (int off = 128; off > 0; off >>= 1) {
    if (tid < off) {
      r0[tid] = fminf(r0[tid], r0[tid + off]);
      r1[tid] = fminf(r1[tid], r1[tid + off]);
      r2[tid] = fminf(r2[tid], r2[tid + off]);
      r3[tid] = fmaxf(r3[tid], r3[tid + off]);
      r4[tid] = fmaxf(r4[tid], r4[tid + off]);
      r5[tid] = fmaxf(r5[tid], r5[tid + off]);
    }
    __syncthreads();
  }
  __shared__ float s_cx, s_cy, s_cz, s_inv;
  __shared__ int s_ok;
  if (tid == 0) {
    float sz = fmaxf(fmaxf(r3[0] - r0[0], r4[0] - r1[0]), r5[0] - r2[0]);
    s_ok = (r3[0] >= r0[0]) && (sz > 0.f);
    s_cx = (r0[0] + r3[0]) * 0.5f;
    s_cy = (r1[0] + r4[0]) * 0.5f;
    s_cz = (r2[0] + r5[0]) * 0.5f;
    s_inv = s_ok ? (1.0f / sz) : 0.f;
  }
  __syncthreads();

  // voxel index + weight per point; count accumulation
  for (int i = tid; i < 1024; i += 256) {
    int w = 0, flat = 0;
    bool valid = (px[i] != 0.f) && (py[i] != 0.f) && (pz[i] != 0.f);
    float p0 = (px[i] - s_cx) * s_inv + 0.5f;
    float p1 = (py[i] - s_cy) * s_inv + 0.5f;
    float p2 = (pz[i] - s_cz) * s_inv + 0.5f;
    bool inb = p0 >= 0.f && p0 <= 1.f && p1 >= 0.f && p1 <= 1.f &&
               p2 >= 0.f && p2 <= 1.f;
    if (s_ok && valid && inb) {
      int g0 = min(max((int)(p0 * 31.f), 0), 31);
      int g1 = min(max((int)(p1 * 31.f), 0), 31);
      int g2 = min(max((int)(p2 * 31.f), 0), 31);
      flat = g0 * 1024 + g1 * 32 + g2;
      w = 1;
    }
    sflat[i] = flat; sw[i] = (unsigned char)w;
    if (w) atomicAdd(&cntbuf[bs * 32768 + flat], 1.0f);
  }
  __syncthreads();

  // scatter features: 256 threads = 2 points x 128 channels per pass
  const float* f2p = f2 + (size_t)bs * CCH * 1024;
  float* gp = g + (size_t)bs * 32768 * CCH;
  int cc = tid & 127;
  int p0i = tid >> 7;
  for (int p = p0i; p < 1024; p += 2) {
    if (sw[p]) {
      float v = f2p[cc * 1024 + p];           // roi_feat[point p][channel cc]
      atomicAdd(&gp[(size_t)sflat[p] * CCH + cc], v);
    }
  }
}

// ------------------------------------------------------------- g /= count
__global__ void k_norm(float* __restrict__ g, const float* __restrict__ cntbuf, long n) {
  long i = (long)blockIdx.x * blockDim.x + threadIdx.x;
  long stride = (long)gridDim.x * blockDim.x;
  for (; i < n; i += stride) {
    float cnt = cntbuf[i >> 7];               // [bs*32768 + flat]
    if (cnt > 0.f) g[i] = g[i] / cnt;
  }
}

extern "C" void kernel_launch(void* const* d_in, const int* in_sizes, int n_in,
                              void* d_out, int out_size, void* d_ws, size_t ws_size,
                              hipStream_t stream) {
  const float*         feat   = (const float*)d_in[0];         // (2,128,120,160)
  const float*         depths = (const float*)d_in[1];         // (2,480,640)
  const float*         intr   = (const float*)d_in[2];         // (2,3,3)
  const unsigned char* masks  = (const unsigned char*)d_in[3]; // (2,8,480,640) bool

  float* out = (float*)d_out;
  float* f2  = out;                    // (2,8,128,32,32) = 2,097,152
  float* g   = out + 2097152;          // (2,8,32,32,32,128) = 67,108,864

  char* ws = (char*)d_ws;
  int*           meta   = (int*)(ws + WS_META_OFF);
  unsigned char* maskq  = (unsigned char*)(ws + WS_MASKQ_OFF);
  float*         cntbuf = (float*)(ws + WS_CNT_OFF);

  k_zero_f4<<<4096, 256, 0, stream>>>((float4*)out, (long)out_size / 4);
  k_zero_f4<<<512, 256, 0, stream>>>((float4*)cntbuf, (16L * 32768) / 4);
  k_meta<<<NB * SSEG, 256, 0, stream>>>(masks, depths, meta, maskq);
  k_resize<<<NB * SSEG * CCH, 128, 0, stream>>>(feat, meta, maskq, f2);
  k_pixelate<<<NB * SSEG, 256, 0, stream>>>(depths, intr, meta, maskq, f2, g, cntbuf);
  k_norm<<<4096, 256, 0, stream>>>(g, cntbuf, 67108864L);
}